// wvfn_75522704933251
// MI455X (gfx1250) — compile-verified
//
#include <hip/hip_runtime.h>
#include <stdint.h>

#define BLOCK 256
// Bytes of walker data staged per block: 256 walkers * 6 floats * 4 B = 6144 B.
#define STAGE_BYTES (BLOCK * 6 * 4)

__global__ __launch_bounds__(BLOCK) void wvfn_75522704933251_kernel(
    const float* __restrict__ Rs,      // [W, 2, 3]
    const float* __restrict__ a,       // [1]
    const float* __restrict__ Cr,      // [2,2,3,2,2]; C[0,0,0] = first 4 elems
    const float* __restrict__ Ci,      // [2,2,3,2,2]
    float* __restrict__ out,           // [2W complex hammy][2W complex psi2] as floats
    int W)
{
    __shared__ __align__(16) float smem[BLOCK * 6];

    const int t = threadIdx.x;
    const int blockStart = blockIdx.x * BLOCK;
    const bool full = (blockStart + BLOCK) <= W;

    float x0, y0, z0, x1, y1, z1;

    if (full) {
        // ---- CDNA5 async global->LDS staging (ASYNCcnt path) ----
        // LDS layout == global layout of this block's 6144-byte slab.
        // Each lane issues 3 x b64 async copies (8 B apart per lane, 2048 B
        // apart per copy) => fully coalesced, bypasses VGPRs.
        const char* gbase = (const char*)(Rs) + (size_t)blockStart * 24u;
        uint64_t gaddr = (uint64_t)(uintptr_t)gbase + (uint32_t)(t * 8);
        // Low 32 bits of a generic pointer to LDS are the LDS byte address.
        uint32_t laddr = (uint32_t)(uintptr_t)(&smem[0]) + (uint32_t)(t * 8);

        #pragma unroll
        for (int k = 0; k < 3; ++k) {
            uint32_t l = laddr + (uint32_t)(k * 2048);
            uint64_t g = gaddr + (uint64_t)(k * 2048);
            asm volatile("global_load_async_to_lds_b64 %0, %1, off"
                         :: "v"(l), "v"(g) : "memory");
        }
        asm volatile("s_wait_asynccnt 0x0" ::: "memory");
        __syncthreads();

        // Conflict-free LDS readback: word index 6*t + j, 6*t injective mod 64
        // over a wave32 -> no bank conflicts.
        const float2* s2 = (const float2*)(&smem[t * 6]);
        float2 v0 = s2[0], v1 = s2[1], v2 = s2[2];
        x0 = v0.x; y0 = v0.y; z0 = v1.x;
        x1 = v1.y; y1 = v2.x; z1 = v2.y;
    } else {
        const int w = blockStart + t;
        if (w >= W) return;                 // uniform-per-block branch: safe
        const float2* g2 = (const float2*)(Rs + (size_t)w * 6u);
        float2 v0 = g2[0], v1 = g2[1], v2 = g2[2];
        x0 = v0.x; y0 = v0.y; z0 = v1.x;
        x1 = v1.y; y1 = v2.x; z1 = v2.y;
    }

    const int w = blockStart + t;

    // Uniform scalars.
    const float a0   = a[0];
    const float cr00 = Cr[0], cr01 = Cr[1], cr10 = Cr[2], cr11 = Cr[3];
    const float ci00 = Ci[0], ci01 = Ci[1], ci10 = Ci[2], ci11 = Ci[3];

    const float Z    = 1.0f / a0;
    // Z^{3/2} / sqrt(pi)
    const float norm = Z * sqrtf(Z) * 0.5641895835477563f;

    const float r0 = sqrtf(x0 * x0 + y0 * y0 + z0 * z0);
    const float r1 = sqrtf(x1 * x1 + y1 * y1 + z1 * z1);

    const float p0 = norm * expf(-Z * r0);
    const float p1 = norm * expf(-Z * r1);

    const float p00 = p0 * p0, p01 = p0 * p1, p11 = p1 * p1;
    const float crs = cr01 + cr10, cis = ci01 + ci10;

    // psi = sum_ab pair[a,b] * C00[a,b]
    const float psi_re = p00 * cr00 + p01 * crs + p11 * cr11;
    const float psi_im = p00 * ci00 + p01 * cis + p11 * ci11;

    // lap_coef = Z^2 - 2Z/r ; nabla_psi = sum_ab (lap_a+lap_b) pair C00
    const float Z2   = Z * Z;
    const float lap0 = Z2 - 2.0f * Z / r0;
    const float lap1 = Z2 - 2.0f * Z / r1;
    const float t00  = 2.0f * lap0 * p00;
    const float t01  = (lap0 + lap1) * p01;
    const float t11  = 2.0f * lap1 * p11;
    const float nab_re = t00 * cr00 + t01 * crs + t11 * cr11;
    const float nab_im = t00 * ci00 + t01 * cis + t11 * ci11;

    // V = -VB / |R0 - R1|, VB = 1.0
    const float dx = x0 - x1, dy = y0 - y1, dz = z0 - z1;
    const float d12 = sqrtf(dx * dx + dy * dy + dz * dz);
    const float V = -1.0f / d12;

    // H_psi = -0.5 * nabla_psi + V * psi
    const float Hr = -0.5f * nab_re + V * psi_re;
    const float Hi = -0.5f * nab_im + V * psi_im;

    // hammy = conj(psi) * H_psi ; psi2 = |psi|^2 (+ 0i)
    const float2 ham = make_float2(psi_re * Hr + psi_im * Hi,
                                   psi_re * Hi - psi_im * Hr);
    const float2 p2  = make_float2(psi_re * psi_re + psi_im * psi_im, 0.0f);

    float2* o = (float2*)out;                 // 2W float2's total
    o[w]            = ham;                    // hammy[w]
    o[(size_t)W + w] = p2;                    // psi2[w]
}

extern "C" void kernel_launch(void* const* d_in, const int* in_sizes, int n_in,
                              void* d_out, int out_size, void* d_ws, size_t ws_size,
                              hipStream_t stream) {
    (void)n_in; (void)out_size; (void)d_ws; (void)ws_size;
    const float* Rs = (const float*)d_in[0];
    const float* a  = (const float*)d_in[1];
    const float* Cr = (const float*)d_in[2];
    const float* Ci = (const float*)d_in[3];

    const int W = in_sizes[0] / 6;            // [W, 2, 3] floats
    const int grid = (W + BLOCK - 1) / BLOCK;

    hipLaunchKernelGGL(wvfn_75522704933251_kernel,
                       dim3(grid), dim3(BLOCK), 0, stream,
                       Rs, a, Cr, Ci, (float*)d_out, W);
}